// gaussianMixtureModel_60936995996103
// MI455X (gfx1250) — compile-verified
//
#include <hip/hip_runtime.h>
#include <math.h>

// CDNA5 / gfx1250, wave32. Full-fp32 WMMA (V_WMMA_F32_16X16X4_F32) to keep
// the reference's fp32 numerics (exp(S) with S ~ -180 cannot tolerate bf16 A/B).

typedef float v2f __attribute__((ext_vector_type(2)));
typedef float v8f __attribute__((ext_vector_type(8)));

#define BDIM   2048
#define DDIM   128
#define KDIM   512
#define TWO_D  256          // GEMM contraction length (2*D)
#define NK4    64           // TWO_D / 4
#define NTILES 32           // KDIM / 16

// workspace layout (float offsets)
#define WS_LSE 0
#define WS_LPI 64
#define WS_K1  (WS_LPI + KDIM)
#define WS_K2  (WS_K1 + KDIM)
#define WS_WS  2048                    // packed W_s : TWO_D*KDIM floats
#define WS_WT  (WS_WS + TWO_D*KDIM)   // packed W_t : TWO_D*KDIM floats

// Pack W[row R in 0..255][col k in 0..511] into per-lane WMMA-B fragment order:
// for a 16x16x4 step, B VGPR j / lane l holds (K = j + 2*(l>=16), N = l&15).
// Packed so each lane's two floats are contiguous -> one global_load_b64/lane.
__device__ __forceinline__ int packIdx(int R, int k) {
    int K4 = R >> 2, r = R & 3;
    int lane = ((r >> 1) << 4) | (k & 15);
    int NT = k >> 4;
    return ((K4 * NTILES + NT) * 32 + lane) * 2 + (r & 1);
}

// ---------------- kernel 0: logsumexp(cluster_prior) -> ws[WS_LSE] ----------
__global__ void gmm_lse_kernel(const float* __restrict__ cp, float* __restrict__ ws) {
    __shared__ float sm[8];
    int tid = threadIdx.x;                       // 256 threads
    float x0 = cp[tid], x1 = cp[tid + 256];
    float m = fmaxf(x0, x1);
    for (int mask = 1; mask < 32; mask <<= 1) m = fmaxf(m, __shfl_xor(m, mask, 32));
    if ((tid & 31) == 0) sm[tid >> 5] = m;
    __syncthreads();
    float M = sm[0];
    for (int i = 1; i < 8; ++i) M = fmaxf(M, sm[i]);
    __syncthreads();
    float s = expf(x0 - M) + expf(x1 - M);
    for (int mask = 1; mask < 32; mask <<= 1) s += __shfl_xor(s, mask, 32);
    if ((tid & 31) == 0) sm[tid >> 5] = s;
    __syncthreads();
    if (tid == 0) {
        float tot = 0.0f;
        for (int i = 0; i < 8; ++i) tot += sm[i];
        ws[WS_LSE] = M + logf(tot);
    }
}

// ---------------- kernel 1: per-(d,k) weights + per-k constants -------------
// one block per k (512 blocks x 128 threads, thread = d)
__global__ void gmm_prep_kernel(const float* __restrict__ cmean,
                                const float* __restrict__ cvsu,
                                const float* __restrict__ cp,
                                float* __restrict__ ws) {
    int k = blockIdx.x, d = threadIdx.x;
    float cm  = cmean[d * KDIM + k];
    float cvu = cvsu[d * KDIM + k];
    float iv  = 1.0f + expf(-cvu);         // 1/var  (var = sigmoid(cvu))
    float var = 1.0f / iv;

    ws[WS_WS + packIdx(d,        k)] = -0.5f * iv;      // pairs with z^2
    ws[WS_WS + packIdx(DDIM + d, k)] = cm * iv;         // pairs with z
    ws[WS_WT + packIdx(d,        k)] = iv;              // pairs with exp(zlv)+zm^2
    ws[WS_WT + packIdx(DDIM + d, k)] = -2.0f * cm * iv; // pairs with zm

    float slv = logf(var);
    float sc  = cm * cm * iv;
    for (int mask = 1; mask < 32; mask <<= 1) {
        slv += __shfl_xor(slv, mask, 32);
        sc  += __shfl_xor(sc,  mask, 32);
    }
    __shared__ float r1[4], r2[4];
    if ((d & 31) == 0) { r1[d >> 5] = slv; r2[d >> 5] = sc; }
    __syncthreads();
    if (d == 0) {
        float SLV = r1[0] + r1[1] + r1[2] + r1[3];
        float SC  = r2[0] + r2[1] + r2[2] + r2[3];
        float lpi = cp[k] - ws[WS_LSE];
        const float LOG2PI = 1.8378770664093453f;
        ws[WS_LPI + k] = lpi;
        ws[WS_K1  + k] = DDIM * lpi - 0.5f * (DDIM * LOG2PI + SLV) - 0.5f * SC;
        ws[WS_K2  + k] = SLV + SC;
    }
}

// ---------------- kernel 2: fused dual GEMM + epilogue ----------------------
// 128 blocks x 256 threads (8 waves). Block = 16 batch rows x all 512 clusters.
// Wave w owns N-columns [w*64, w*64+64) as 4 WMMA tiles; K-loop over 256 in 4s.
#define ASTR 260   // padded LDS row stride (floats) to spread banks

__global__ void gmm_main_kernel(const float* __restrict__ zmean,
                                const float* __restrict__ zlvq,
                                const float* __restrict__ zin,
                                const float* __restrict__ ws,
                                float* __restrict__ out) {
    __shared__ float As[16 * ASTR];
    __shared__ float At[16 * ASTR];
    __shared__ float red[16][4];
    __shared__ float f4[16];
    __shared__ float zinv[16];

    int tid = threadIdx.x;
    int b0  = blockIdx.x * 16;

    if (tid < 64) ((float*)red)[tid] = 0.0f;

    // ---- phase 0: build A panels in LDS, accumulate fourth = 0.5*sum(zlv+1)
    {
        int r = tid >> 4, c = tid & 15;          // half-wave per row
        float fsum = 0.0f;
        #pragma unroll
        for (int j = 0; j < 8; ++j) {
            int d  = c + j * 16;
            int gi = (b0 + r) * DDIM + d;
            float zm = zmean[gi], zl = zlvq[gi], zz = zin[gi];
            As[r * ASTR + d]        = zz * zz;
            As[r * ASTR + DDIM + d] = zz;
            At[r * ASTR + d]        = expf(zl) + zm * zm;
            At[r * ASTR + DDIM + d] = zm;
            fsum += 0.5f * (zl + 1.0f);
        }
        for (int mask = 1; mask < 16; mask <<= 1) fsum += __shfl_xor(fsum, mask, 32);
        if (c == 0) f4[r] = fsum;
    }
    __syncthreads();

    int lane = tid & 31, wv = tid >> 5;
    int lhi = lane >> 4, llo = lane & 15;

    const float* K1v = ws + WS_K1;
    const float* K2v = ws + WS_K2;
    const float* lpi = ws + WS_LPI;
    const v2f* Ws2 = (const v2f*)(ws + WS_WS);
    const v2f* Wt2 = (const v2f*)(ws + WS_WT);

    v8f accS[4], accT[4];
    #pragma unroll
    for (int t = 0; t < 4; ++t) {
        float k1 = K1v[wv * 64 + t * 16 + llo];   // C/D: column N = lane&15
        float k2 = K2v[wv * 64 + t * 16 + llo];
        #pragma unroll
        for (int v = 0; v < 8; ++v) { accS[t][v] = k1; accT[t][v] = k2; }
    }

    // ---- GEMM S: S = K1 + A_s * W_s  (fp32 WMMA, 16x16x4)
    for (int K4 = 0; K4 < NK4; ++K4) {
        int kk = K4 * 4;
        v2f a = *(const v2f*)&As[llo * ASTR + kk + 2 * lhi];  // A frag
        #pragma unroll
        for (int t = 0; t < 4; ++t) {
            v2f b = Ws2[(K4 * NTILES + (wv * 4 + t)) * 32 + lane];
            accS[t] = __builtin_amdgcn_wmma_f32_16x16x4_f32(
                false, a, false, b, (short)0, accS[t], false, false);
        }
    }
    // ---- GEMM T: T = K2 + A_t * W_t
    for (int K4 = 0; K4 < NK4; ++K4) {
        int kk = K4 * 4;
        v2f a = *(const v2f*)&At[llo * ASTR + kk + 2 * lhi];
        #pragma unroll
        for (int t = 0; t < 4; ++t) {
            v2f b = Wt2[(K4 * NTILES + (wv * 4 + t)) * 32 + lane];
            accT[t] = __builtin_amdgcn_wmma_f32_16x16x4_f32(
                false, a, false, b, (short)0, accT[t], false, false);
        }
    }

    // ---- epilogue: pu = exp(S)+1e-10; row sums of {pu, pu*T, pu*ln pu, pu*lnpi}
    float lp[4];
    #pragma unroll
    for (int t = 0; t < 4; ++t) lp[t] = lpi[wv * 64 + t * 16 + llo];

    #pragma unroll
    for (int v = 0; v < 8; ++v) {           // C/D: M = v + 8*(lane>=16)
        float q0 = 0.f, q1 = 0.f, q2 = 0.f, q3 = 0.f;
        #pragma unroll
        for (int t = 0; t < 4; ++t) {
            float pu = expf(accS[t][v]) + 1e-10f;
            accS[t][v] = pu;
            float Tv = accT[t][v];
            q0 += pu; q1 += pu * Tv; q2 += pu * logf(pu); q3 += pu * lp[t];
        }
        for (int mask = 1; mask < 16; mask <<= 1) {   // reduce 16-lane half
            q0 += __shfl_xor(q0, mask, 32); q1 += __shfl_xor(q1, mask, 32);
            q2 += __shfl_xor(q2, mask, 32); q3 += __shfl_xor(q3, mask, 32);
        }
        if (llo == 0) {
            int M = v + 8 * lhi;
            atomicAdd(&red[M][0], q0); atomicAdd(&red[M][1], q1);
            atomicAdd(&red[M][2], q2); atomicAdd(&red[M][3], q3);
        }
    }
    __syncthreads();

    if (tid < 16) {
        float Z  = red[tid][0];
        float iZ = 1.0f / Z;
        float lZ = logf(Z);
        float second = 0.5f * red[tid][1] * iZ;
        float tmp6   = red[tid][2] * iZ - lZ;     // sum P log P
        float tmp7   = red[tid][3] * iZ;          // sum P log pi
        float third  = tmp7 - tmp6;
        out[(size_t)BDIM * KDIM + b0 + tid] = second - third - f4[tid]; // neg_loss
        zinv[tid] = iZ;
    }
    __syncthreads();

    // ---- P = pu / Z
    #pragma unroll
    for (int t = 0; t < 4; ++t) {
        #pragma unroll
        for (int v = 0; v < 8; ++v) {
            int M = v + 8 * lhi;
            out[(size_t)(b0 + M) * KDIM + wv * 64 + t * 16 + llo] = accS[t][v] * zinv[M];
        }
    }
}

extern "C" void kernel_launch(void* const* d_in, const int* in_sizes, int n_in,
                              void* d_out, int out_size, void* d_ws, size_t ws_size,
                              hipStream_t stream) {
    (void)in_sizes; (void)n_in; (void)out_size; (void)ws_size;
    const float* z_mean = (const float*)d_in[0];
    const float* zlv    = (const float*)d_in[1];
    const float* z      = (const float*)d_in[2];
    const float* cmean  = (const float*)d_in[3];
    const float* cvsu   = (const float*)d_in[4];
    const float* cprior = (const float*)d_in[5];
    float* ws  = (float*)d_ws;
    float* out = (float*)d_out;

    gmm_lse_kernel <<<1,        256, 0, stream>>>(cprior, ws);
    gmm_prep_kernel<<<KDIM,     DDIM, 0, stream>>>(cmean, cvsu, cprior, ws);
    gmm_main_kernel<<<BDIM/16,  256, 0, stream>>>(z_mean, zlv, z, ws, out);
}